// CTCLoss_76312978915692
// MI455X (gfx1250) — compile-verified
//
#include <hip/hip_runtime.h>
#include <hip/hip_bf16.h>
#include <cstdint>

#define NEGV (-1e30f)

__device__ __forceinline__ float lae(float x, float y) {
    // logaddexp with finite -1e30 sentinel (matches jnp.logaddexp on finite inputs)
    float m = fmaxf(x, y);
    float d = fminf(x, y) - m;
    return m + log1pf(expf(d));
}

// Async DMA: 16 bytes per lane from global -> LDS (ASYNCcnt-tracked, CDNA5).
__device__ __forceinline__ void async_load16(uint32_t lds_byte_off, const void* gsrc) {
    uint64_t ga = (uint64_t)(uintptr_t)gsrc;
    asm volatile("global_load_async_to_lds_b128 %0, %1, off"
                 :: "v"(lds_byte_off), "v"(ga)
                 : "memory");
}

__device__ __forceinline__ void wait_async0() {
    asm volatile("s_wait_asynccnt 0" ::: "memory");
}

// One workgroup per batch element; one thread per extended-label position.
// LDS layout: A[L2+2] | B[L2+2] | rows[2 chunks][4 rows][V]
__global__ void ctc_alpha_kernel(const float* __restrict__ log_probs,
                                 const int*   __restrict__ in_len,
                                 const int*   __restrict__ targets,
                                 const int*   __restrict__ tgt_len,
                                 float*       __restrict__ per_batch,
                                 int T, int V, int S)
{
    extern __shared__ float smem[];
    const int b   = blockIdx.x;
    const int tid = threadIdx.x;
    const int L2  = 2 * S + 1;

    float* A   = smem;               // alpha buffer 0 (2-slot NEG padding at front)
    float* Bf  = A + (L2 + 2);       // alpha buffer 1
    float* row = Bf + (L2 + 2);      // 2 chunks x 4 rows x V floats

    const int Lb    = tgt_len[b];
    const int inLen = in_len[b];
    const float* lp = log_probs + (size_t)b * (size_t)T * (size_t)V;
    const int*   tg = targets + (size_t)b * (size_t)S;

    const uint32_t rowBytes   = (uint32_t)V * 4u;       // 512 B per row
    const uint32_t chunkBytes = 4u * rowBytes;          // 2 KB per chunk
    const uint32_t rowbase    = (uint32_t)(uintptr_t)(void*)row;  // LDS byte offset

    // ---- issue chunk 0 (rows 1..4) into buf0 while we set up ----
    if (tid < 32) {
        #pragma unroll
        for (int j = 0; j < 4; ++j) {
            int r = 1 + j;
            if (r < T)
                async_load16(rowbase + (uint32_t)j * rowBytes + (uint32_t)tid * 16u,
                             lp + (size_t)r * V + tid * 4);
        }
    }

    // ---- per-thread extended-label state ----
    const int s = tid;
    int  ext_s = 0;
    bool skip = false, valid = false;
    float a0 = NEGV;                                    // register-resident own alpha
    if (s < L2) {
        ext_s = (s & 1) ? tg[(s - 1) >> 1] : 0;
        int ext_m2 = (s >= 2) ? (((s - 2) & 1) ? tg[(s - 3) >> 1] : 0) : 0;
        skip  = (s >= 2) && (s & 1) && (ext_s != ext_m2);
        valid = s < (2 * Lb + 1);
        if (s == 0)      a0 = lp[0];                    // log_probs[b,0,blank]
        else if (s == 1) a0 = (Lb > 0) ? lp[ext_s] : NEGV;
        A[2 + s] = a0;
    }
    if (tid == 0) { A[0] = NEGV; A[1] = NEGV; Bf[0] = NEGV; Bf[1] = NEGV; }

    if (tid < 32) wait_async0();                        // chunk 0 resident
    __syncthreads();

    float* cur = A;
    float* nxt = Bf;

    // ---- time scan: chunks of 4 rows, one chunk DMA in flight ----
    int t = 1;
    for (int kc = 0; t < T; ++kc) {
        const float* rbuf = row + (size_t)(kc & 1) * 4 * V;

        // issue next chunk into the other buffer (its previous contents were
        // fully consumed before the chunk-end barrier of chunk kc-1)
        if (tid < 32) {
            const uint32_t dst = rowbase + (uint32_t)((kc + 1) & 1) * chunkBytes;
            #pragma unroll
            for (int j = 0; j < 4; ++j) {
                int r = 1 + 4 * (kc + 1) + j;
                if (r < T)
                    async_load16(dst + (uint32_t)j * rowBytes + (uint32_t)tid * 16u,
                                 lp + (size_t)r * V + tid * 4);
            }
        }

        #pragma unroll
        for (int j = 0; j < 4; ++j) {
            if (t >= T) break;
            const float* rp = rbuf + j * V;
            const bool alive = (t < inLen);
            if (s < L2) {
                float a1 = cur[1 + s];                  // alpha[s-1]
                float a2 = cur[s];                      // alpha[s-2]
                float r  = lae(a0, a1);
                float r2 = lae(r, a2);
                r = skip ? r2 : r;                      // branchless skip transition
                float nv = valid ? (r + rp[ext_s]) : NEGV;
                a0 = alive ? nv : a0;                   // freeze after input end
                nxt[2 + s] = a0;
            }
            ++t;
            float* tmp = cur; cur = nxt; nxt = tmp;
            if (j < 3 && t < T) __syncthreads();        // chunk-end barrier handles j==3/tail
        }

        if (tid < 32) wait_async0();                    // next chunk landed in LDS
        __syncthreads();                                // alpha + rows visible to all waves
    }

    if (tid == 0) {
        const int L = Lb;
        float a_blank = cur[2 + 2 * L];
        int   i0      = (2 * L - 1 > 0) ? (2 * L - 1) : 0;
        float a_label = cur[2 + i0];
        float ll   = lae(a_blank, (L > 0) ? a_label : NEGV);
        float loss = -ll;
        if (loss > 1e29f) loss = 0.0f;                  // zero_infinity
        int Lc = (L > 0) ? L : 1;
        per_batch[b] = loss / (float)Lc;
    }
}

// Deterministic reduction: sequential sum by a single lane (bit-stable across replays).
__global__ void ctc_reduce_kernel(const float* __restrict__ per_batch,
                                  float* __restrict__ out, int B)
{
    if (blockIdx.x == 0 && threadIdx.x == 0) {
        float acc = 0.0f;
        for (int i = 0; i < B; ++i) acc += per_batch[i];
        out[0] = acc / (float)B;
    }
}

extern "C" void kernel_launch(void* const* d_in, const int* in_sizes, int n_in,
                              void* d_out, int out_size, void* d_ws, size_t ws_size,
                              hipStream_t stream) {
    const float* log_probs = (const float*)d_in[0];
    const int*   lp_len    = (const int*)d_in[1];
    const int*   targets   = (const int*)d_in[2];
    const int*   tgt_len   = (const int*)d_in[3];

    const int B  = in_sizes[1];                 // 64
    const int S  = in_sizes[2] / B;             // 256
    const int V  = 128;                         // vocab size from reference head
    const int T  = (in_sizes[0] / B) / V;       // 1024
    const int L2 = 2 * S + 1;                   // 513

    float* per_batch = (float*)d_ws;

    int block = (L2 + 31) & ~31;                // 544: one thread per extended label
    if (block > 1024) block = 1024;
    size_t shmem = (size_t)(2 * (L2 + 2) + 8 * V) * sizeof(float);

    ctc_alpha_kernel<<<B, block, shmem, stream>>>(log_probs, lp_len, targets, tgt_len,
                                                  per_batch, T, V, S);
    ctc_reduce_kernel<<<1, 32, 0, stream>>>(per_batch, (float*)d_out, B);
}